// DeformAttnFusionMultiImage_71184787964532
// MI455X (gfx1250) — compile-verified
//
#include <hip/hip_runtime.h>
#include <hip/hip_bf16.h>

// DeformAttnFusionMultiImage for MI455X (gfx1250, wave32).
// Only camera 5 matters (the scan carry is overwritten each iteration).
// All dense 16-channel projections run on V_WMMA_F32_16X16X4_F32.

#define CC    16
#define NPTS  400000
#define DCAM  6
#define FHH   256
#define FWW   448
#define PIX   (FHH * FWW)     // 114688
#define EPSBN 1e-3f

typedef __attribute__((ext_vector_type(2))) float v2f;
typedef __attribute__((ext_vector_type(8))) float v8f;

static __device__ __forceinline__ v8f wmma4(v2f a, v2f b, v8f c) {
  // D = A(16x4,f32) * B(4x16,f32) + C(16x16,f32)
  return __builtin_amdgcn_wmma_f32_16x16x4_f32(
      /*neg_a=*/false, a, /*neg_b=*/false, b,
      /*c_mod=*/(short)0, c, /*reuse_a=*/false, /*reuse_b=*/false);
}

// eval-mode BatchNorm1d folded to per-channel affine: y = sc*x + sh
static __device__ __forceinline__ void bn_coef(const float* __restrict__ bnp,
                                               int n, float& sc, float& sh) {
  float g  = bnp[n];
  float be = bnp[16 + n];
  float mu = bnp[32 + n];
  float vr = bnp[48 + n];
  sc = g * rsqrtf(vr + EPSBN);
  sh = be - mu * sc;
}

// ---------------------------------------------------------------------------
// Kernel 1: value projection for camera 5.
//   val[b][pix][c'] = sum_c fpn[b][5][c][pix] * w_val[c'][c] + b_val[c']
// One wave per 16-pixel tile. 2*PIX/16 = 14336 waves -> 1792 blocks of 8.
// ---------------------------------------------------------------------------
__global__ __launch_bounds__(256) void valproj_kernel(
    const float* __restrict__ fpn, const float* __restrict__ w_val,
    const float* __restrict__ b_val, float* __restrict__ val)
{
  const int lane  = threadIdx.x & 31;
  const int wave  = threadIdx.x >> 5;
  const int wg    = blockIdx.x * 8 + wave;
  const int b     = wg / (PIX / 16);
  const int pix0  = (wg % (PIX / 16)) * 16;
  const int m     = lane & 15;       // A row (pixel) / B-D column (channel)
  const int hi16  = lane >> 4;       // lane half selects K/M split
  const int khalf = 2 * hi16;

  const float* src = fpn + (size_t)(b * DCAM + 5) * CC * PIX;

  v8f acc;
  const float bias = b_val[m];
#pragma unroll
  for (int r = 0; r < 8; ++r) acc[r] = bias;

#pragma unroll
  for (int j = 0; j < 4; ++j) {
    const int k = 4 * j + khalf;
    v2f a, bf;
    a.x  = src[(size_t)k * PIX + pix0 + m];         // A[m][k] (channel-strided)
    a.y  = src[(size_t)(k + 1) * PIX + pix0 + m];
    bf.x = w_val[m * CC + k];                        // B[k][n] = w_val[n][k]
    bf.y = w_val[m * CC + k + 1];
    acc  = wmma4(a, bf, acc);
  }

  float* dst = val + ((size_t)b * PIX + pix0) * CC;
#pragma unroll
  for (int r = 0; r < 8; ++r)
    dst[(r + 8 * hi16) * CC + m] = acc[r];           // D[M=r+8*half][N=m]
}

// ---------------------------------------------------------------------------
// Kernel 2: per-point pipeline. One wave per 16-point tile.
// 400000/16 = 25000 tiles -> 3125 blocks of 8 waves (no tail).
// ---------------------------------------------------------------------------
__global__ __launch_bounds__(256) void deform_main_kernel(
    const float* __restrict__ pts,  const float* __restrict__ proj,
    const float* __restrict__ w_key, const float* __restrict__ b_key, const float* __restrict__ bn_key,
    const float* __restrict__ w_tr,  const float* __restrict__ b_tr,  const float* __restrict__ bn_tr,
    const float* __restrict__ w_off, const float* __restrict__ b_off,
    const float* __restrict__ w_att, const float* __restrict__ b_att,
    const float* __restrict__ w_fuse, const float* __restrict__ b_fuse, const float* __restrict__ bn_fuse,
    const float* __restrict__ val,  float* __restrict__ out)
{
  __shared__ float s_org[8][16][16];   // pts_org tile
  __shared__ float s_off[8][16][32];   // 32 offset outputs
  __shared__ float s_att[8][16][16];   // attention logits
  __shared__ float s_fuse[8][16][32];  // relu(concat(pts_pre, img_feats))

  const int lane  = threadIdx.x & 31;
  const int wave  = threadIdx.x >> 5;
  const int tile  = blockIdx.x * 8 + wave;
  const int base  = tile * 16;
  const int m     = lane & 15;
  const int hi16  = lane >> 4;
  const int khalf = 2 * hi16;

  float (*org)[16]   = s_org[wave];
  float (*offb)[32]  = s_off[wave];
  float (*attb)[16]  = s_att[wave];
  float (*fuseb)[32] = s_fuse[wave];

  // ---- GEMM 1: pts_org = BN(X @ w_key^T + b_key) ----
  v2f a[4];
  {
    const float* row = pts + (size_t)(base + m) * CC;
#pragma unroll
    for (int j = 0; j < 4; ++j) {
      const int k = 4 * j + khalf;
      a[j].x = row[k];
      a[j].y = row[k + 1];
    }
  }
  {
    v8f acc;
    const float bias = b_key[m];
#pragma unroll
    for (int r = 0; r < 8; ++r) acc[r] = bias;
#pragma unroll
    for (int j = 0; j < 4; ++j) {
      const int k = 4 * j + khalf;
      v2f bf;
      bf.x = w_key[m * CC + k];
      bf.y = w_key[m * CC + k + 1];
      acc  = wmma4(a[j], bf, acc);
    }
    float sc, sh;
    bn_coef(bn_key, m, sc, sh);
#pragma unroll
    for (int r = 0; r < 8; ++r)
      org[r + 8 * hi16][m] = acc[r] * sc + sh;
  }
  __syncthreads();

  // Re-fragment pts_org as A-matrix from LDS.
#pragma unroll
  for (int j = 0; j < 4; ++j) {
    const int k = 4 * j + khalf;
    a[j].x = org[m][k];
    a[j].y = org[m][k + 1];
  }

  // ---- GEMM 2: pts_pre = BN(pts_org @ w_tr^T + b_tr); relu into fuseb[:,0:16]
  {
    v8f acc;
    const float bias = b_tr[m];
#pragma unroll
    for (int r = 0; r < 8; ++r) acc[r] = bias;
#pragma unroll
    for (int j = 0; j < 4; ++j) {
      const int k = 4 * j + khalf;
      v2f bf;
      bf.x = w_tr[m * CC + k];
      bf.y = w_tr[m * CC + k + 1];
      acc  = wmma4(a[j], bf, acc);
    }
    float sc, sh;
    bn_coef(bn_tr, m, sc, sh);
#pragma unroll
    for (int r = 0; r < 8; ++r)
      fuseb[r + 8 * hi16][m] = fmaxf(acc[r] * sc + sh, 0.0f);
  }

  // ---- GEMM 3: offsets (32 outputs -> two 16-wide WMMA accumulators) ----
  {
    v8f lo, hi;
    const float blo = b_off[m], bhi = b_off[16 + m];
#pragma unroll
    for (int r = 0; r < 8; ++r) { lo[r] = blo; hi[r] = bhi; }
#pragma unroll
    for (int j = 0; j < 4; ++j) {
      const int k = 4 * j + khalf;
      v2f bl, bh;
      bl.x = w_off[m * CC + k];
      bl.y = w_off[m * CC + k + 1];
      bh.x = w_off[(16 + m) * CC + k];
      bh.y = w_off[(16 + m) * CC + k + 1];
      lo   = wmma4(a[j], bl, lo);
      hi   = wmma4(a[j], bh, hi);
    }
#pragma unroll
    for (int r = 0; r < 8; ++r) {
      offb[r + 8 * hi16][m]      = lo[r];
      offb[r + 8 * hi16][16 + m] = hi[r];
    }
  }

  // ---- GEMM 4: attention logits ----
  {
    v8f acc;
    const float bias = b_att[m];
#pragma unroll
    for (int r = 0; r < 8; ++r) acc[r] = bias;
#pragma unroll
    for (int j = 0; j < 4; ++j) {
      const int k = 4 * j + khalf;
      v2f bf;
      bf.x = w_att[m * CC + k];
      bf.y = w_att[m * CC + k + 1];
      acc  = wmma4(a[j], bf, acc);
    }
#pragma unroll
    for (int r = 0; r < 8; ++r)
      attb[r + 8 * hi16][m] = acc[r];
  }
  __syncthreads();

  // ---- softmax + deformable bilinear sampling (camera 5 only) ----
  // Each lane owns point m and heads {hi16, hi16+2}.
  {
    const int gi = base + m;
    const float* pc = proj + ((size_t)5 * NPTS + gi) * 4;
    const int   bidx  = (int)pc[0];
    const float xf    = pc[1];
    const float yf    = pc[2];
    const float depth = pc[3];
    const int xi = (int)xf;   // trunc toward zero == torch .long()
    const int yi = (int)yf;
    const bool valid = (xi >= 0) && (xi < FWW) && (yi >= 0) && (yi < FHH) &&
                       (depth >= 0.0f);
    const float* valb = val + (size_t)bidx * PIX * CC;

#pragma unroll
    for (int t = 0; t < 2; ++t) {
      const int h = hi16 + 2 * t;
      // softmax over the 4 sampling points of this head
      float l0 = attb[m][h * 4 + 0];
      float l1 = attb[m][h * 4 + 1];
      float l2 = attb[m][h * 4 + 2];
      float l3 = attb[m][h * 4 + 3];
      float mx = fmaxf(fmaxf(l0, l1), fmaxf(l2, l3));
      float e0 = __expf(l0 - mx), e1 = __expf(l1 - mx);
      float e2 = __expf(l2 - mx), e3 = __expf(l3 - mx);
      float inv = 1.0f / (e0 + e1 + e2 + e3);
      float aw[4] = {e0 * inv, e1 * inv, e2 * inv, e3 * inv};

      float f0 = 0.f, f1 = 0.f, f2 = 0.f, f3 = 0.f;
#pragma unroll
      for (int p = 0; p < 4; ++p) {
        const float ox = offb[m][(h * 4 + p) * 2 + 0];
        const float oy = offb[m][(h * 4 + p) * 2 + 1];
        // loc.x*FW - 0.5 == xi + ox - 0.5 (the /norm * fw cancels)
        const float px = (float)xi + ox - 0.5f;
        const float py = (float)yi + oy - 0.5f;
        const float fx0 = floorf(px), fy0 = floorf(py);
        const int   x0  = (int)fx0,   y0  = (int)fy0;
        const float wx1 = px - fx0,   wy1 = py - fy0;
        const float wx0 = 1.f - wx1,  wy0 = 1.f - wy1;
        const float w   = aw[p];
        const float cw[4] = {w * wx0 * wy0, w * wx1 * wy0,
                             w * wx0 * wy1, w * wx1 * wy1};
        const int cx[4] = {x0, x0 + 1, x0,     x0 + 1};
        const int cy[4] = {y0, y0,     y0 + 1, y0 + 1};
#pragma unroll
        for (int c = 0; c < 4; ++c) {
          if (cx[c] >= 0 && cx[c] < FWW && cy[c] >= 0 && cy[c] < FHH) {
            const float4 v4 = *(const float4*)(
                valb + ((size_t)cy[c] * FWW + cx[c]) * CC + h * 4);
            f0 += cw[c] * v4.x;
            f1 += cw[c] * v4.y;
            f2 += cw[c] * v4.z;
            f3 += cw[c] * v4.w;
          }
        }
      }
      if (valid) {
        fuseb[m][16 + h * 4 + 0] = fmaxf(f0, 0.f);
        fuseb[m][16 + h * 4 + 1] = fmaxf(f1, 0.f);
        fuseb[m][16 + h * 4 + 2] = fmaxf(f2, 0.f);
        fuseb[m][16 + h * 4 + 3] = fmaxf(f3, 0.f);
      } else {  // invalid points revert to pts_org
        fuseb[m][16 + h * 4 + 0] = fmaxf(org[m][h * 4 + 0], 0.f);
        fuseb[m][16 + h * 4 + 1] = fmaxf(org[m][h * 4 + 1], 0.f);
        fuseb[m][16 + h * 4 + 2] = fmaxf(org[m][h * 4 + 2], 0.f);
        fuseb[m][16 + h * 4 + 3] = fmaxf(org[m][h * 4 + 3], 0.f);
      }
    }
  }
  __syncthreads();

  // ---- GEMM 5: out = relu(BN(fused(16x32) @ w_fuse^T + b_fuse)) ----
  {
    v8f acc;
    const float bias = b_fuse[m];
#pragma unroll
    for (int r = 0; r < 8; ++r) acc[r] = bias;
#pragma unroll
    for (int j = 0; j < 8; ++j) {            // K = 32 -> 8 chunks of 4
      const int k = 4 * j + khalf;
      v2f aa, bb;
      aa.x = fuseb[m][k];
      aa.y = fuseb[m][k + 1];
      bb.x = w_fuse[m * 32 + k];             // B[k][n] = w_fuse[n][k]
      bb.y = w_fuse[m * 32 + k + 1];
      acc  = wmma4(aa, bb, acc);
    }
    float sc, sh;
    bn_coef(bn_fuse, m, sc, sh);
    float* orow = out + (size_t)base * CC;
#pragma unroll
    for (int r = 0; r < 8; ++r)
      orow[(r + 8 * hi16) * CC + m] = fmaxf(acc[r] * sc + sh, 0.0f);
  }
}

extern "C" void kernel_launch(void* const* d_in, const int* in_sizes, int n_in,
                              void* d_out, int out_size, void* d_ws, size_t ws_size,
                              hipStream_t stream) {
  const float* pts     = (const float*)d_in[0];
  const float* fpn     = (const float*)d_in[1];
  const float* proj    = (const float*)d_in[2];
  // d_in[3] image_mask: unused by the forward pass
  const float* w_key   = (const float*)d_in[4];
  const float* b_key   = (const float*)d_in[5];
  const float* bn_key  = (const float*)d_in[6];
  const float* w_tr    = (const float*)d_in[7];
  const float* b_tr    = (const float*)d_in[8];
  const float* bn_tr   = (const float*)d_in[9];
  const float* w_off   = (const float*)d_in[10];
  const float* b_off   = (const float*)d_in[11];
  const float* w_att   = (const float*)d_in[12];
  const float* b_att   = (const float*)d_in[13];
  const float* w_val   = (const float*)d_in[14];
  const float* b_val   = (const float*)d_in[15];
  const float* w_fuse  = (const float*)d_in[16];
  const float* b_fuse  = (const float*)d_in[17];
  const float* bn_fuse = (const float*)d_in[18];

  float* val = (float*)d_ws;  // 2 * PIX * 16 floats = 14.68 MB scratch

  // Camera-5 value projection: 14336 pixel-tiles / 8 waves per block.
  valproj_kernel<<<1792, 256, 0, stream>>>(fpn, w_val, b_val, val);

  // Point pipeline: 25000 point-tiles / 8 waves per block.
  deform_main_kernel<<<3125, 256, 0, stream>>>(
      pts, proj, w_key, b_key, bn_key, w_tr, b_tr, bn_tr,
      w_off, b_off, w_att, b_att, w_fuse, b_fuse, bn_fuse,
      val, (float*)d_out);
}